// AttnGreedySearchV2_1700807049359
// MI455X (gfx1250) — compile-verified
//
#include <hip/hip_runtime.h>
#include <math.h>

typedef float v2f __attribute__((ext_vector_type(2)));
typedef float v8f __attribute__((ext_vector_type(8)));

#define BS      65536
#define IN_DIM  100
#define HID     16
#define NCORP   10
#define ROWS    (BS * NCORP)      // 655360 rows in flattened projection GEMM
#define WPB     8                 // waves per block (256 threads, wave32)

// ---------------------------------------------------------------------------
// Kernel 1: proj = reshape(item_corpus,(ROWS,100)) @ W_proj + b_proj
//           tvec = tanh(proj @ W_t)
// One wave owns a 16-row x 16-col tile; K handled by V_WMMA_F32_16X16X4_F32.
//
// fp32 WMMA layouts (ISA 7.12.2):
//   A 16x4 : lane l, vgpr v -> A[m = l&15][k = v + 2*(l>>4)]
//   B 4x16 : lane l, vgpr v -> B[k = v + 2*(l>>4)][n = l&15]
//   C/D    : lane l, vgpr r -> D[m = r + 8*(l>>4)][n = l&15]
// ---------------------------------------------------------------------------
__global__ __launch_bounds__(256) void proj_tvec_kernel(
    const float* __restrict__ item_corpus,  // (ROWS, 100)
    const float* __restrict__ W_proj,       // (100, 16)
    const float* __restrict__ b_proj,       // (16)
    const float* __restrict__ W_t,          // (16, 16)
    float* __restrict__ proj,               // (ROWS, 16)
    float* __restrict__ tvec)               // (ROWS, 16)
{
    __shared__ float lds[WPB][16][17];      // padded rows: conflict-light relayout

    const int lane  = threadIdx.x & 31;
    const int wv    = threadIdx.x >> 5;
    const int l15   = lane & 15;            // A-row / B-col / D-col of this lane
    const int khalf = (lane >> 4) * 2;      // K offset of this lane's half
    const int rofs  = (lane >> 4) * 8;      // D-row offset of this lane's half

    const long tile = (long)blockIdx.x * WPB + wv;
    const long row0 = tile * 16;

    // ---- GEMM 1: (16 x 100) @ (100 x 16), K in 25 steps of 4 ----
    v8f acc = {};
    const float* arow = item_corpus + (row0 + l15) * IN_DIM;
    #pragma unroll
    for (int k0 = 0; k0 < IN_DIM; k0 += 4) {
        v2f a = *(const v2f*)(arow + k0 + khalf);           // A[m][k..k+1]
        v2f b;
        b.x = W_proj[(k0 + khalf + 0) * HID + l15];         // B[k][n]
        b.y = W_proj[(k0 + khalf + 1) * HID + l15];
        acc = __builtin_amdgcn_wmma_f32_16x16x4_f32(
                  false, a, false, b, (short)0, acc, false, false);
    }

    // bias: D column of every accumulator element is l15
    const float bias = b_proj[l15];
    #pragma unroll
    for (int r = 0; r < 8; ++r) acc[r] += bias;

    // store proj and stage into LDS for D->A relayout
    #pragma unroll
    for (int r = 0; r < 8; ++r) {
        proj[(row0 + r + rofs) * HID + l15] = acc[r];
        lds[wv][r + rofs][l15] = acc[r];
    }
    __syncthreads();

    // ---- GEMM 2: (16 x 16) @ (16 x 16), K in 4 steps of 4, then tanh ----
    v8f acc2 = {};
    #pragma unroll
    for (int kc = 0; kc < 4; ++kc) {
        v2f a;
        a.x = lds[wv][l15][kc * 4 + khalf + 0];
        a.y = lds[wv][l15][kc * 4 + khalf + 1];
        v2f b;
        b.x = W_t[(kc * 4 + khalf + 0) * HID + l15];
        b.y = W_t[(kc * 4 + khalf + 1) * HID + l15];
        acc2 = __builtin_amdgcn_wmma_f32_16x16x4_f32(
                   false, a, false, b, (short)0, acc2, false, false);
    }
    #pragma unroll
    for (int r = 0; r < 8; ++r)
        tvec[(row0 + r + rofs) * HID + l15] = tanhf(acc2[r]);
}

// ---------------------------------------------------------------------------
// Kernel 2: greedy attention search, one thread per batch element.
// softmax dropped (monotone -> argmax of logits); strict '>' keeps the
// first-max index like jnp.argmax.
// ---------------------------------------------------------------------------
__global__ __launch_bounds__(256) void greedy_kernel(
    const float* __restrict__ user_intent,  // (BS, 16)
    const float* __restrict__ proj,         // (BS*10, 16)
    const float* __restrict__ tvec,         // (BS*10, 16)
    const float* __restrict__ W_s,          // (16, 16)
    const int*   __restrict__ search_num_p, // scalar
    float* __restrict__ out)                // (BS, search_num+1, 16)
{
    __shared__ float Ws[256];
    Ws[threadIdx.x] = W_s[threadIdx.x];
    __syncthreads();

    const int  b     = blockIdx.x * blockDim.x + threadIdx.x;
    const int  steps = *search_num_p;
    const long obase = (long)b * (steps + 1) * HID;

    float sum[HID];
    #pragma unroll
    for (int h = 0; h < HID; ++h) {
        float v = user_intent[(long)b * HID + h];
        sum[h] = v;
        out[obase + h] = v;
    }

    const float* tb = tvec + (long)b * NCORP * HID;
    const float* cb = proj + (long)b * NCORP * HID;

    for (int it = 0; it < steps; ++it) {
        const float inv = 1.0f / (float)(it + 1);

        float s[HID];
        #pragma unroll
        for (int h = 0; h < HID; ++h) {
            float a = 0.0f;
            #pragma unroll
            for (int hp = 0; hp < HID; ++hp)
                a += (sum[hp] * inv) * Ws[hp * HID + h];
            s[h] = tanhf(a);
        }

        float best = -__builtin_inff();
        int bidx = 0;
        for (int c = 0; c < NCORP; ++c) {
            float sc = 0.0f;
            #pragma unroll
            for (int h = 0; h < HID; ++h)
                sc += tb[c * HID + h] * s[h];
            if (sc > best) { best = sc; bidx = c; }
        }

        #pragma unroll
        for (int h = 0; h < HID; ++h) {
            float v = cb[bidx * HID + h];
            sum[h] += v;
            out[obase + (long)(it + 1) * HID + h] = v;
        }
    }
}

extern "C" void kernel_launch(void* const* d_in, const int* in_sizes, int n_in,
                              void* d_out, int out_size, void* d_ws, size_t ws_size,
                              hipStream_t stream) {
    const float* user_intent = (const float*)d_in[0];
    const float* item_corpus = (const float*)d_in[1];
    const float* W_proj      = (const float*)d_in[2];
    const float* b_proj      = (const float*)d_in[3];
    const float* W_s         = (const float*)d_in[4];
    const float* W_t         = (const float*)d_in[5];
    const int*   search_num  = (const int*)d_in[6];
    float* out = (float*)d_out;

    float* proj = (float*)d_ws;                       // ROWS*16 floats
    float* tv   = proj + (size_t)ROWS * HID;          // ROWS*16 floats

    const int tiles   = ROWS / 16;                    // 40960 wave tiles
    const int blocks1 = tiles / WPB;                  // 5120 blocks x 8 waves

    proj_tvec_kernel<<<blocks1, 256, 0, stream>>>(item_corpus, W_proj, b_proj,
                                                  W_t, proj, tv);
    greedy_kernel<<<BS / 256, 256, 0, stream>>>(user_intent, proj, tv,
                                                W_s, search_num, out);
}